// DynamicFilterLayer_42288247996640
// MI455X (gfx1250) — compile-verified
//
#include <hip/hip_runtime.h>
#include <cstdint>

// ---------------------------------------------------------------------------
// DynamicFilterLayer on MI455X (gfx1250, wave32)
//
// Per instance (64): x:(100000,10) -> Dense(8)+ReLU -> Dense(8)+ReLU ->
//                    Dense(1) -> sigmoid.
//
// Bandwidth floor: 256MB points + 25.6MB out @ 23.3 TB/s ~= 12us.
// Layers 1 and 2 run on the matrix pipe via V_WMMA_F32_16X16X4_F32
// (full fp32 precision), layer 3 + sigmoid on VALU.
// ---------------------------------------------------------------------------

typedef __attribute__((ext_vector_type(2))) float v2f;
typedef __attribute__((ext_vector_type(8))) float v8f;

#define N_MASK   100000
#define IN_CH    10
#define F1       8
#define F2       8
#define TILE     32          // points per wave-iteration (2 WMMA M-tiles)
#define WAVES    8           // waves per block (256 threads)
#define CPB      40          // chunks (blocks) per instance
#define NTILES   3125        // 100000 / 32, exact
#define TPB      ((NTILES + CPB - 1) / CPB)
#define LSTRIDE  18          // LDS row stride in floats: even (8B aligned b64)
                             // and odd/2 -> conflict-free across 16/32 lanes

// filters layout per instance (169 floats):
//  [0,80)    W1 (8x10 row-major)
//  [80,88)   b1
//  [88,152)  W2 (8x8 row-major)
//  [152,160) b2
//  [160,168) W3 (1x8)
//  [168]     b3
#define OFF_W1 0
#define OFF_B1 80
#define OFF_W2 88
#define OFF_B2 152
#define OFF_W3 160
#define OFF_B3 168
#define FILT_STRIDE 169

__global__ __launch_bounds__(256) void dynfilter_wmma_kernel(
    const float* __restrict__ points,
    const float* __restrict__ filters,
    float* __restrict__ out)
{
    const int inst  = blockIdx.x / CPB;       // uniform per block
    const int chunk = blockIdx.x % CPB;
    const int lane  = threadIdx.x & 31;
    const int wave  = threadIdx.x >> 5;
    const int f     = lane & 15;              // N index (filter) for B/C/D
    const int grp   = lane >> 4;              // K-half selector for A/B frags

    __shared__ float lds[WAVES][TILE * LSTRIDE];
    float* sbuf = lds[wave];

    const float* W = filters + (size_t)inst * FILT_STRIDE;
    const float* P = points  + (size_t)inst * N_MASK * IN_CH;
    float*       O = out     + (size_t)inst * N_MASK;

    // ---- per-block weight fragments ------------------------------------
    // B fragment (4x16 f32): lane holds N=f; lanes 0-15: K=kb,kb+1 in .x/.y,
    // lanes 16-31: K=kb+2,kb+3. Columns f>=8 and K>=10 are zero padding.
    v2f B1[3];
#pragma unroll
    for (int j = 0; j < 3; ++j) {
        const int kb = j * 4;
        const int ks = kb + grp * 2;
        float bx = 0.0f, by = 0.0f;
        if (f < F1) {
            if (ks     < IN_CH) bx = W[OFF_W1 + f * IN_CH + ks];
            if (ks + 1 < IN_CH) by = W[OFF_W1 + f * IN_CH + ks + 1];
        }
        B1[j].x = bx; B1[j].y = by;
    }
    const float bias1 = (f < F1) ? W[OFF_B1 + f] : 0.0f;

    v2f B2[2];
#pragma unroll
    for (int j = 0; j < 2; ++j) {
        const int ks = j * 4 + grp * 2;
        float bx = 0.0f, by = 0.0f;
        if (f < F2) {
            bx = W[OFF_W2 + f * F1 + ks];
            by = W[OFF_W2 + f * F1 + ks + 1];
        }
        B2[j].x = bx; B2[j].y = by;
    }
    const float bias2 = (f < F2) ? W[OFF_B2 + f] : 0.0f;

    float w3[8];
#pragma unroll
    for (int j = 0; j < 8; ++j) w3[j] = W[OFF_W3 + j];   // uniform -> SGPRs
    const float b3 = W[OFF_B3];

    // ---- tile loop ------------------------------------------------------
    const int t0 = chunk * TPB;
    int tEnd = t0 + TPB; if (tEnd > NTILES) tEnd = NTILES;

    for (int t = t0 + wave; t < tEnd; t += WAVES) {
        const int p0 = t * TILE;

        // prefetch this wave's next tile into the cache hierarchy
        const int tn = t + WAVES;
        if (tn < tEnd)
            __builtin_prefetch(P + (size_t)(tn * TILE + lane) * IN_CH, 0, 0);

        // ---------------- layer 1: (16x10) x (10x8) via 3 K-chunks ------
        v8f acc[2] = {};
#pragma unroll
        for (int T = 0; T < 2; ++T) {
            // A fragment source: this lane owns point M = f of M-tile T.
            const float* pbase = P + (size_t)(p0 + T * 16 + f) * IN_CH;
#pragma unroll
            for (int j = 0; j < 3; ++j) {
                const int kb = j * 4;
                v2f a;
                if (kb == 8 && grp == 1) {
                    a.x = 0.0f; a.y = 0.0f;              // K=10,11 pad
                } else {
                    a = *(const v2f*)(pbase + kb + grp * 2);  // 8B aligned
                }
                acc[T] = __builtin_amdgcn_wmma_f32_16x16x4_f32(
                    false, a, false, B1[j], (short)0, acc[T], false, false);
            }
        }

        // bias + ReLU, transpose into LDS: row = point-in-tile, col = chan
#pragma unroll
        for (int T = 0; T < 2; ++T)
#pragma unroll
            for (int v = 0; v < 8; ++v) {
                float x = acc[T][v] + bias1;             // C/D: M=grp*8+v, N=f
                x = fmaxf(x, 0.0f);
                sbuf[(T * 16 + grp * 8 + v) * LSTRIDE + f] = x;
            }
        __builtin_amdgcn_wave_barrier();
        asm volatile("s_wait_dscnt 0" ::: "memory");

        // ---------------- layer 2: (16x8) x (8x8) via 2 K-chunks --------
        v8f acc2[2] = {};
#pragma unroll
        for (int T = 0; T < 2; ++T) {
#pragma unroll
            for (int j = 0; j < 2; ++j) {
                const int ks = j * 4 + grp * 2;
                v2f a = *(const v2f*)&sbuf[(T * 16 + f) * LSTRIDE + ks];
                acc2[T] = __builtin_amdgcn_wmma_f32_16x16x4_f32(
                    false, a, false, B2[j], (short)0, acc2[T], false, false);
            }
        }
        __builtin_amdgcn_wave_barrier();

        // bias + ReLU, transpose back into LDS (same rows; LDS is in-order
        // per wave, so reads above complete before these stores)
#pragma unroll
        for (int T = 0; T < 2; ++T)
#pragma unroll
            for (int v = 0; v < 8; ++v) {
                float x = acc2[T][v] + bias2;
                x = fmaxf(x, 0.0f);
                sbuf[(T * 16 + grp * 8 + v) * LSTRIDE + f] = x;
            }
        __builtin_amdgcn_wave_barrier();
        asm volatile("s_wait_dscnt 0" ::: "memory");

        // ---------------- layer 3 + sigmoid: one point per lane ---------
        const float* row = &sbuf[lane * LSTRIDE];
        float s = b3;
#pragma unroll
        for (int j = 0; j < 8; ++j) s = fmaf(row[j], w3[j], s);
        const float sig = 1.0f / (1.0f + __expf(-s));
        O[p0 + lane] = sig;                              // coalesced 128B
        __builtin_amdgcn_wave_barrier();
    }
}

extern "C" void kernel_launch(void* const* d_in, const int* in_sizes, int n_in,
                              void* d_out, int out_size, void* d_ws, size_t ws_size,
                              hipStream_t stream) {
    const float* points  = (const float*)d_in[0];
    const float* filters = (const float*)d_in[1];
    float*       out     = (float*)d_out;

    dim3 grid(64 * CPB);   // 2560 blocks
    dim3 block(256);       // 8 waves, wave32
    hipLaunchKernelGGL(dynfilter_wmma_kernel, grid, block, 0, stream,
                       points, filters, out);
}